// NEGLoss_73555609912003
// MI455X (gfx1250) — compile-verified
//
#include <hip/hip_runtime.h>

// NEG loss (negative-sampling NLL) for MI455X / gfx1250, wave32.
//
// Optimized pipeline: inverse-CDF categorical sampling (O(V) build + O(log V)
// per draw) replaces the O(V)-per-draw exponential race -> ~5e4x less sampler
// work; the whole loss becomes launch-overhead bound (~1 MB total HBM traffic,
// CDF fully L2-resident).
//
//   1. init:       weights[V]=0, pnum/pden=0
//   2. scanlocal:  q[v]=distr[v]^0.75; per-block LDS inclusive scan -> cdf,
//                  block sums -> bsum (scan's own last element, bit-consistent)
//   3. scanblocks: one block scans bsum -> exclusive boff, writes total
//   4. addoff:     cdf[v] += boff[block]
//   5. sample:     one thread per (neg,b): hash -> u, binary search cdf,
//                  deterministic rejection of the positive (<=8 attempts),
//                  atomicAdd(weights[sample],1); threads j<B also add the
//                  positive count (fused histogram)
//   6. dot:        block partials num = sum w_t*input[i,t], den = sum w_t
//   7. finalize:   one wave32; v_wmma_f32_16x16x4_f32 cross-lane reduction;
//                  out = -num/den
//
// Workspace floats: cdf[V] | weights[V] | pnum[64] | pden[64] |
//                   bsum[512] | boff[512] | total[1]

#define NUM_NEG  5
#define NPART    64
#define NBLK_MAX 512   // room for 512 block sums; scan_blocks (256 threads) handles V <= 256*256 = 65536

typedef __attribute__((ext_vector_type(2))) float v2f;
typedef __attribute__((ext_vector_type(8))) float v8f;

__global__ void neg_init_kernel(float* __restrict__ buf, int n) {
  int i = blockIdx.x * blockDim.x + threadIdx.x;
  int stride = gridDim.x * blockDim.x;
  for (; i < n; i += stride) buf[i] = 0.0f;
}

// Per-block inclusive scan of q = distr^0.75 (Hillis-Steele in LDS).
__global__ __launch_bounds__(256) void neg_scan_local_kernel(
    const float* __restrict__ distr, float* __restrict__ cdf,
    float* __restrict__ bsum, int V) {
  __shared__ float s[256];
  const int tid = threadIdx.x;
  const int v = blockIdx.x * 256 + tid;
  float x = 0.0f;
  if (v < V) {
    float d = distr[v];                      // > 0 by construction
    x = __expf(0.75f * __logf(d));           // distr^0.75 (unnormalized is fine)
  }
  s[tid] = x;
  __syncthreads();
#pragma unroll
  for (int off = 1; off < 256; off <<= 1) {
    float t = (tid >= off) ? s[tid - off] : 0.0f;
    __syncthreads();
    s[tid] += t;
    __syncthreads();
  }
  if (v < V) cdf[v] = s[tid];
  if (tid == 255) bsum[blockIdx.x] = s[255]; // same association as the scan
}

// Single-block scan of per-block sums -> exclusive offsets + grand total.
__global__ __launch_bounds__(256) void neg_scan_blocks_kernel(
    const float* __restrict__ bsum, float* __restrict__ boff,
    float* __restrict__ total, int nblk) {
  __shared__ float s[256];
  const int tid = threadIdx.x;
  float x = (tid < nblk) ? bsum[tid] : 0.0f;
  s[tid] = x;
  __syncthreads();
#pragma unroll
  for (int off = 1; off < 256; off <<= 1) {
    float t = (tid >= off) ? s[tid - off] : 0.0f;
    __syncthreads();
    s[tid] += t;
    __syncthreads();
  }
  if (tid < nblk) boff[tid] = s[tid] - x;    // exclusive
  if (tid == nblk - 1) total[0] = s[tid];
}

__global__ __launch_bounds__(256) void neg_addoff_kernel(
    float* __restrict__ cdf, const float* __restrict__ boff, int V) {
  int v = blockIdx.x * 256 + threadIdx.x;
  if (v < V) cdf[v] += boff[blockIdx.x];
}

// One thread per (neg_sample, row). CDF (200 KB) is L2-resident (192 MB L2).
__global__ __launch_bounds__(256) void neg_sample_kernel(
    const float* __restrict__ cdf, const float* __restrict__ total_p,
    const int* __restrict__ target, float* __restrict__ weights, int B, int V) {
  const int j = blockIdx.x * 256 + threadIdx.x;
  const int njobs = NUM_NEG * B;
  if (j >= njobs) return;
  const int b = j % B;
  const int tgt = target[b];
  if (j < B) atomicAdd(&weights[tgt], 1.0f); // fused positive-count histogram
  const float total = total_p[0];
  int v = tgt;
#pragma unroll 1
  for (unsigned int a = 0; a < 8u && v == tgt; ++a) { // rejection of positive
    // stateless 32-bit hash RNG, counter = 8*j + attempt
    unsigned int h = ((unsigned int)j * 8u + a) * 0x9E3779B9u + 0x2545F491u;
    h ^= h >> 16; h *= 0x7FEB352Du;
    h ^= h >> 15; h *= 0x846CA68Bu;
    h ^= h >> 16;
    float u = (float)(h >> 8) * (1.0f / 16777216.0f); // [0,1)
    float x = u * total;
    int lo = 0, hi = V - 1;                  // smallest v with cdf[v] > x
    while (lo < hi) {
      int mid = (lo + hi) >> 1;
      if (cdf[mid] > x) hi = mid; else lo = mid + 1;
    }
    v = lo;
  }
  if (v == tgt) v = (tgt + 1 < V) ? tgt + 1 : 0; // deterministic fallback
  atomicAdd(&weights[v], 1.0f);
}

__global__ __launch_bounds__(256) void neg_dot_kernel(
    const float* __restrict__ input, const int* __restrict__ target,
    const float* __restrict__ weights, float* __restrict__ pnum,
    float* __restrict__ pden, int B, int V) {
  __shared__ float sn[256], sd[256];
  const int tid = threadIdx.x;
  float n = 0.0f, d = 0.0f;
  for (int i = blockIdx.x * 256 + tid; i < B; i += gridDim.x * 256) {
    int t = target[i];
    float w = weights[t];
    float x = input[(size_t)i * (size_t)V + (size_t)t]; // sparse gather
    n = fmaf(w, x, n);
    d += w;
  }
  sn[tid] = n; sd[tid] = d;
  __syncthreads();
  for (int s = 128; s > 0; s >>= 1) {
    if (tid < s) { sn[tid] += sn[tid + s]; sd[tid] += sd[tid + s]; }
    __syncthreads();
  }
  if (tid == 0) { pnum[blockIdx.x] = sn[0]; pden[blockIdx.x] = sd[0]; }
}

// One wave32; no divergence before the WMMA (EXEC must be all 1s).
// A (16x4 f32, 2 VGPRs/lane): lane m<16 holds row m K={0,1}; lane m+16 K={2,3}.
//   num goes in the even-K slot, den in the odd-K slot of every lane.
// B (4x16 f32): selector, even K -> column N=0, odd K -> column N=1.
// D = A*B: column 0 = 16 row-sums of num, column 1 = 16 row-sums of den.
// D layout: lanes 0-15 hold N=lane (VGPR r -> M=r), lanes 16-31 N=lane-16 (M=r+8).
__global__ __launch_bounds__(32) void neg_finalize_wmma_kernel(
    const float* __restrict__ pnum, const float* __restrict__ pden,
    float* __restrict__ out) {
  const int lane = threadIdx.x;
  float num = 0.0f, den = 0.0f;
#pragma unroll
  for (int k = 0; k < NPART; k += 32) {      // uniform trip count, no divergence
    num += pnum[lane + k];
    den += pden[lane + k];
  }
  v2f a; a.x = num; a.y = den;
  const int l16 = lane & 15;
  v2f bm;
  bm.x = (l16 == 0) ? 1.0f : 0.0f;           // even K feeds column N=0
  bm.y = (l16 == 1) ? 1.0f : 0.0f;           // odd  K feeds column N=1
  v8f c = {};
  // (neg_a, A, neg_b, B, c_mod, C, reuse_a, reuse_b)
  c = __builtin_amdgcn_wmma_f32_16x16x4_f32(false, a, false, bm, (short)0, c,
                                            false, false);
  float s = c[0] + c[1] + c[2] + c[3] + c[4] + c[5] + c[6] + c[7];
  float n0 = __shfl(s, 0, 32), n16 = __shfl(s, 16, 32); // num halves (N=0)
  float d1 = __shfl(s, 1, 32), d17 = __shfl(s, 17, 32); // den halves (N=1)
  if (lane == 0) out[0] = -(n0 + n16) / (d1 + d17);
}

extern "C" void kernel_launch(void* const* d_in, const int* in_sizes, int n_in,
                              void* d_out, int out_size, void* d_ws,
                              size_t ws_size, hipStream_t stream) {
  const float* input  = (const float*)d_in[0];  // [B, V] fp32
  const int*   target = (const int*)d_in[1];    // [B] int
  const float* distr  = (const float*)d_in[2];  // [V] fp32
  const int B = in_sizes[1];
  const int V = in_sizes[2];
  const int nblk = (V + 255) / 256;             // 197 for V=50257 (<= NBLK_MAX)

  float* ws      = (float*)d_ws;
  float* cdf     = ws;                          // [V]
  float* weights = cdf + V;                     // [V]
  float* pnum    = weights + V;                 // [NPART]
  float* pden    = pnum + NPART;                // [NPART]
  float* bsum    = pden + NPART;                // [NBLK_MAX]
  float* boff    = bsum + NBLK_MAX;             // [NBLK_MAX]
  float* total   = boff + NBLK_MAX;             // [1]

  // zero weights + pnum + pden (contiguous V + 2*NPART floats)
  neg_init_kernel<<<(V + 2 * NPART + 255) / 256, 256, 0, stream>>>(
      weights, V + 2 * NPART);
  neg_scan_local_kernel<<<nblk, 256, 0, stream>>>(distr, cdf, bsum, V);
  neg_scan_blocks_kernel<<<1, 256, 0, stream>>>(bsum, boff, total, nblk);
  neg_addoff_kernel<<<nblk, 256, 0, stream>>>(cdf, boff, V);
  neg_sample_kernel<<<(NUM_NEG * B + 255) / 256, 256, 0, stream>>>(
      cdf, total, target, weights, B, V);
  neg_dot_kernel<<<NPART, 256, 0, stream>>>(input, target, weights, pnum, pden,
                                            B, V);
  neg_finalize_wmma_kernel<<<1, 32, 0, stream>>>(pnum, pden, (float*)d_out);
}